// BBoxModel_21689584845415
// MI455X (gfx1250) — compile-verified
//
#include <hip/hip_runtime.h>

#define Hc 2048
#define Wc 2048
#define Nn (Hc*Wc)
#define MAXN 100
#define THRv 0.3f
#define BOXTHR 0.7f
#define SIZETHR 5.0f
#define STRIDE_PX 262144   // 1024 blocks * 256 threads

typedef __attribute__((ext_vector_type(2))) float v2f;
typedef __attribute__((ext_vector_type(8))) float v8f;

// order-preserving float <-> int encoding (self-inverse); enc(0.0f)==0
__device__ __forceinline__ int encf(float f){ int b=__float_as_int(f); return b ^ ((b>>31) & 0x7fffffff); }
__device__ __forceinline__ float decf(int e){ int b = e ^ ((e>>31) & 0x7fffffff); return __int_as_float(b); }

// ---------------- labeling ----------------

__global__ void k_init(const float* __restrict__ hot, int* __restrict__ lab, int* __restrict__ lut){
  int p = blockIdx.x*256 + threadIdx.x;
  if (p <= Nn) lut[p] = p;
  if (p <  Nn) lab[p] = (hot[p] > THRv) ? (p+1) : 0;
}

__global__ void k_prop(const int* __restrict__ lab, int* __restrict__ lut){
  __shared__ int tile[10][34];
  int bx = blockIdx.x*32, by = blockIdx.y*8;
  int t = threadIdx.y*32 + threadIdx.x;
  for (int i=t; i<340; i+=256){
    int ly = i/34, lx = i%34;
    int gy = by + ly - 1, gx = bx + lx - 1;
    int v = 0;
    if (gx>=0 && gx<Wc && gy>=0 && gy<Hc) v = lab[gy*Wc+gx];
    tile[ly][lx] = v;
  }
  __syncthreads();
  int lx = threadIdx.x+1, ly = threadIdx.y+1;
  int c = tile[ly][lx];
  int m = c;
  #pragma unroll
  for(int dy=-1;dy<=1;dy++)
    #pragma unroll
    for(int dx=-1;dx<=1;dx++){ int v=tile[ly+dy][lx+dx]; m = v>m ? v : m; }
  if (c>0 && m>c) atomicMax(&lut[c], m);
}

__global__ void k_compress(int* __restrict__ lut){
  int i = blockIdx.x*256 + threadIdx.x;
  if (i > Nn) return;
  int v = lut[i];
  if (v == i) return;
  int n = lut[v];
  while (n != v) { v = n; n = lut[v]; }   // monotone chains: in-place chase is race-safe
  lut[i] = v;
}

__global__ void k_apply(int* __restrict__ lab, const int* __restrict__ lut){
  int p = blockIdx.x*256 + threadIdx.x;
  if (p >= Nn) return;
  int pf = p + 65536;
  if (pf < Nn) __builtin_prefetch(&lab[pf], 0, 0);   // global_prefetch_b8
  lab[p] = lut[lab[p]];
}

// ---------------- relabel (present + scan of 4M+1) ----------------

__global__ void k_clear(int* __restrict__ scr){
  int i = blockIdx.x*256 + threadIdx.x;
  if (i <= Nn) scr[i] = 0;
}

__global__ void k_present(const int* __restrict__ lab, int* __restrict__ scr){
  int p = blockIdx.x*256 + threadIdx.x;
  if (p < Nn) scr[lab[p]] = 1;
}

__global__ void k_scan1(const int* __restrict__ scr, int* __restrict__ bsum){
  __shared__ int sh[256];
  int base = blockIdx.x*4096 + threadIdx.x*16;
  int s = 0;
  #pragma unroll
  for(int j=0;j<16;j++){ int idx=base+j; if(idx<=Nn) s += scr[idx]; }
  sh[threadIdx.x]=s; __syncthreads();
  for(int off=128; off>0; off>>=1){
    if(threadIdx.x<off) sh[threadIdx.x]+=sh[threadIdx.x+off];
    __syncthreads();
  }
  if(threadIdx.x==0) bsum[blockIdx.x]=sh[0];
}

__global__ void k_scan2(int* __restrict__ bsum, int nb){
  __shared__ int sh[1056];
  for(int i=threadIdx.x;i<nb;i+=256) sh[i]=bsum[i];
  __syncthreads();
  if(threadIdx.x==0){ int run=0; for(int i=0;i<nb;i++){ run+=sh[i]; sh[i]=run; } }
  __syncthreads();
  for(int i=threadIdx.x;i<nb;i+=256) bsum[i]=sh[i];
}

__global__ void k_scan3(int* __restrict__ scr, const int* __restrict__ bsum){
  __shared__ int sh[256];
  int t = threadIdx.x;
  int base = blockIdx.x*4096 + t*16;
  int v[16]; int s=0;
  #pragma unroll
  for(int j=0;j<16;j++){ int idx=base+j; int x=(idx<=Nn)?scr[idx]:0; v[j]=x; s+=x; }
  sh[t]=s; __syncthreads();
  for(int off=1; off<256; off<<=1){
    int x = (t>=off)? sh[t-off] : 0;
    __syncthreads();
    sh[t]+=x;
    __syncthreads();
  }
  int excl = sh[t]-s;
  int run = ((blockIdx.x>0)? bsum[blockIdx.x-1] : 0) + excl;
  #pragma unroll
  for(int j=0;j<16;j++){ int idx=base+j; if(idx<=Nn){ run+=v[j]; scr[idx]=run-1; } }
}

// ---------------- moments ----------------

__global__ void k_zero_acc(double* __restrict__ acc){
  int i=threadIdx.x;
  if(i<MAXN*8) acc[i]=0.0;
}

__global__ void k_m1(const float* __restrict__ hot, int* __restrict__ lab,
                     const int* __restrict__ ranks, double* __restrict__ acc){
  __shared__ float sacc[MAXN*4];
  for(int i=threadIdx.x;i<MAXN*4;i+=256) sacc[i]=0.f;
  __syncthreads();
  int tid = blockIdx.x*256+threadIdx.x;
  for(int p=tid; p<Nn; p+=STRIDE_PX){
    int l = ranks[lab[p]];
    if (l>=MAXN) l=0;
    lab[p]=l;
    float x = (float)(p & (Wc-1)), y = (float)(p >> 11);
    atomicAdd(&sacc[l*4+0], 1.0f);
    atomicAdd(&sacc[l*4+1], x);
    atomicAdd(&sacc[l*4+2], y);
    atomicAdd(&sacc[l*4+3], hot[p]);
  }
  __syncthreads();
  for(int i=threadIdx.x;i<MAXN*4;i+=256){
    float v=sacc[i];
    if (v!=0.f) atomicAdd(&acc[(i>>2)*8 + (i&3)], (double)v);
  }
}

__global__ void k_mu(const double* __restrict__ acc, float* __restrict__ stats){
  int c=threadIdx.x; if(c>=MAXN) return;
  double area=acc[c*8+0];
  float mux=0.f,muy=0.f;
  if(area>0.0){ mux=(float)(acc[c*8+1]/area); muy=(float)(acc[c*8+2]/area); }
  stats[c*12+0]=mux; stats[c*12+1]=muy;
  stats[c*12+2]=(float)area; stats[c*12+3]=(float)acc[c*8+3];
}

__global__ void k_m2(const int* __restrict__ lab, const float* __restrict__ stats,
                     double* __restrict__ acc){
  __shared__ float sacc[MAXN*3];
  __shared__ float smu[MAXN*2];
  for(int i=threadIdx.x;i<MAXN*3;i+=256) sacc[i]=0.f;
  for(int i=threadIdx.x;i<MAXN*2;i+=256) smu[i]=stats[(i>>1)*12+(i&1)];
  __syncthreads();
  int tid=blockIdx.x*256+threadIdx.x;
  for(int p=tid;p<Nn;p+=STRIDE_PX){
    int l=lab[p];
    float cx=(float)(p & (Wc-1))-smu[l*2+0];
    float cy=(float)(p >> 11)   -smu[l*2+1];
    atomicAdd(&sacc[l*3+0],cx*cx);
    atomicAdd(&sacc[l*3+1],cx*cy);
    atomicAdd(&sacc[l*3+2],cy*cy);
  }
  __syncthreads();
  for(int i=threadIdx.x;i<MAXN*3;i+=256){
    float v=sacc[i];
    if(v!=0.f) atomicAdd(&acc[(i/3)*8+4+(i%3)],(double)v);
  }
}

__global__ void k_eigen(const double* __restrict__ acc, float* __restrict__ stats,
                        int* __restrict__ mm){
  int t=threadIdx.x;
  for(int i=t;i<MAXN*4;i+=512) mm[i]=0;      // enc(0.0f)==0 -> built-in clamp vs 0
  if(t>=MAXN) return;
  double area=acc[t*8+0];
  float m00=0.f,m01=0.f,m10=0.f,m11=0.f,margin=0.f;
  if(area>0.0){
    float vx =(float)(acc[t*8+4]/area);
    float vxy=(float)(acc[t*8+5]/area);
    float vy =(float)(acc[t*8+6]/area);
    float theta=0.5f*atan2f(2.f*vxy, vx-vy);
    float cth=cosf(theta), sth=sinf(theta);
    m00=cth; m01=-sth; m10=sth; m11=cth;
    float tr=vx+vy;
    float disc=fmaxf((vx-vy)*(vx-vy)+4.f*vxy*vxy, 1e-12f);
    float sq=sqrtf(disc);
    float l2=fmaxf((tr-sq)*0.5f,0.f);
    margin=sqrtf(sqrtf(l2))*4.0f;            // sqrt(sqrt(eig)) * 4 * MAR, per reference
  }
  stats[t*12+4]=m00; stats[t*12+5]=m01; stats[t*12+6]=m10; stats[t*12+7]=m11;
  stats[t*12+8]=margin;
}

__global__ void k_mm(const int* __restrict__ lab, const float* __restrict__ stats,
                     int* __restrict__ mm){
  __shared__ int smm[MAXN*4];
  __shared__ float sm[MAXN*6];
  for(int i=threadIdx.x;i<MAXN*4;i+=256) smm[i]=0;
  for(int i=threadIdx.x;i<MAXN;i+=256){
    sm[i*6+0]=stats[i*12+0]; sm[i*6+1]=stats[i*12+1];
    sm[i*6+2]=stats[i*12+4]; sm[i*6+3]=stats[i*12+5];
    sm[i*6+4]=stats[i*12+6]; sm[i*6+5]=stats[i*12+7];
  }
  __syncthreads();
  int tid=blockIdx.x*256+threadIdx.x;
  for(int p=tid;p<Nn;p+=STRIDE_PX){
    int l=lab[p];
    float cx=(float)(p & (Wc-1))-sm[l*6+0];
    float cy=(float)(p >> 11)   -sm[l*6+1];
    float rx=sm[l*6+2]*cx + sm[l*6+4]*cy;   // rxy = m^T @ coords
    float ry=sm[l*6+3]*cx + sm[l*6+5]*cy;
    atomicMin(&smm[l*4+0], encf(rx));
    atomicMax(&smm[l*4+1], encf(rx));
    atomicMin(&smm[l*4+2], encf(ry));
    atomicMax(&smm[l*4+3], encf(ry));
  }
  __syncthreads();
  for(int i=threadIdx.x;i<MAXN*4;i+=256){
    int v=smm[i];
    if((i&1)==0){ if(v<0) atomicMin(&mm[i],v); }
    else        { if(v>0) atomicMax(&mm[i],v); }
  }
}

// ---------------- output: batched 2x2 @ 2x5 via V_WMMA_F32_16X16X4_F32 ----------------
// Pack 2 components per WMMA: comp j occupies rows {2j,2j+1}, K-block j (K=2j,2j+1),
// columns {5j..5j+4}. Off-diagonal blocks are A-zeros x B-zeros, so diagonal blocks are exact.

__global__ void k_out(const float* __restrict__ stats, const int* __restrict__ mm,
                      const float* __restrict__ scale, float* __restrict__ out){
  __shared__ float fin[MAXN][12];
  int lane = threadIdx.x;
  for(int c=lane;c<MAXN;c+=32){
    float area=stats[c*12+2], level=stats[c*12+3], margin=stats[c*12+8];
    float minx=decf(mm[c*4+0])-margin;
    float maxx=decf(mm[c*4+1])+margin;
    float miny=decf(mm[c*4+2])-margin;
    float maxy=decf(mm[c*4+3])+margin;
    bool msk = (area>0.f) && (level/area > BOXTHR) && (maxx-minx > SIZETHR) && (maxy-miny > SIZETHR);
    if(!(area>0.f)){ minx=maxx=miny=maxy=0.f; }
    fin[c][0]=minx; fin[c][1]=maxx; fin[c][2]=miny; fin[c][3]=maxy;
    fin[c][4]=stats[c*12+4]; fin[c][5]=stats[c*12+5];
    fin[c][6]=stats[c*12+6]; fin[c][7]=stats[c*12+7];
    fin[c][8]=stats[c*12+0]; fin[c][9]=stats[c*12+1];
    fin[c][10]=msk?1.f:0.f;
  }
  __syncthreads();
  float s2 = scale[0]*2.0f;
  int Mrow = lane & 15;        // A: M row / B: column
  int kb   = lane >> 4;        // K-block: .x->K=2kb, .y->K=2kb+1
  for(int g=0; g<MAXN/2; g++){
    v2f a; a.x=0.f; a.y=0.f;
    if (Mrow < 4){
      int cc = Mrow>>1, ii = Mrow&1;
      if (kb == cc){
        int comp = g*2+cc;
        a.x = fin[comp][4 + ii*2 + 0];   // rot[ii][0]
        a.y = fin[comp][4 + ii*2 + 1];   // rot[ii][1]
      }
    }
    v2f b; b.x=0.f; b.y=0.f;
    int colc = Mrow - 5*kb;
    if (colc>=0 && colc<5){
      int comp=g*2+kb;
      float minx=fin[comp][0],maxx=fin[comp][1],miny=fin[comp][2],maxy=fin[comp][3];
      b.x = (colc==0||colc==3||colc==4)? minx : maxx;   // recs row 0
      b.y = (colc==0||colc==1||colc==4)? miny : maxy;   // recs row 1
    }
    v8f cz = {};
    v8f d = __builtin_amdgcn_wmma_f32_16x16x4_f32(false, a, false, b, (short)0, cz, false, false);
    if (lane < 10){
      int cl = lane/5, k = lane%5, comp = g*2+cl;
      float nx = d[2*cl+0] + fin[comp][8];
      float ny = d[2*cl+1] + fin[comp][9];
      bool msk = fin[comp][10] > 0.5f;
      out[comp*10 + k*2 + 0] = msk ? nx*s2 : 0.f;
      out[comp*10 + k*2 + 1] = msk ? ny*s2 : 0.f;
    }
  }
}

// ---------------- host ----------------

extern "C" void kernel_launch(void* const* d_in, const int* in_sizes, int n_in,
                              void* d_out, int out_size, void* d_ws, size_t ws_size,
                              hipStream_t stream) {
  const float* hot   = (const float*)d_in[0];
  const float* scale = (const float*)d_in[1];
  char* ws = (char*)d_ws;

  int* lab = (int*)ws;                                   // N ints
  size_t off = (size_t)Nn*4;
  int* lut = (int*)(ws+off); off += (size_t)(Nn+1)*4;    // N+1 ints
  int* scr = lut;                                        // reuse lut after labeling rounds
  off = (off+7)&~(size_t)7;
  int* bsum = (int*)(ws+off); off += 1056*4;
  off = (off+7)&~(size_t)7;
  double* acc = (double*)(ws+off); off += (size_t)MAXN*8*sizeof(double);
  float* stats = (float*)(ws+off); off += (size_t)MAXN*12*4;
  int* mm = (int*)(ws+off); off += (size_t)MAXN*4*4;
  float* out = (float*)d_out;

  int nbN1 = (Nn+1+255)/256;   // covers indices 0..N
  int nbN  = (Nn+255)/256;

  k_init<<<nbN1,256,0,stream>>>(hot,lab,lut);

  dim3 pg(Wc/32, Hc/8), pb(32,8);
  for(int r=0;r<3;r++){
    k_prop<<<pg,pb,0,stream>>>(lab,lut);
    k_compress<<<nbN1,256,0,stream>>>(lut);
    k_apply<<<nbN,256,0,stream>>>(lab,lut);
  }

  k_clear<<<nbN1,256,0,stream>>>(scr);
  k_present<<<nbN,256,0,stream>>>(lab,scr);

  int NB = (Nn+1+4095)/4096;   // 1025 blocks
  k_scan1<<<NB,256,0,stream>>>(scr,bsum);
  k_scan2<<<1,256,0,stream>>>(bsum,NB);
  k_scan3<<<NB,256,0,stream>>>(scr,bsum);

  k_zero_acc<<<1,1024,0,stream>>>(acc);
  k_m1<<<1024,256,0,stream>>>(hot,lab,scr,acc);
  k_mu<<<1,128,0,stream>>>(acc,stats);
  k_m2<<<1024,256,0,stream>>>(lab,stats,acc);
  k_eigen<<<1,512,0,stream>>>(acc,stats,mm);
  k_mm<<<1024,256,0,stream>>>(lab,stats,mm);
  k_out<<<1,32,0,stream>>>(stats,mm,scale,out);
}